// PostProcessor_44504451121717
// MI455X (gfx1250) — compile-verified
//
#include <hip/hip_runtime.h>

// Deformable post-processor for MI455X (gfx1250, wave32).
// Bandwidth-bound (485 MB streamed, ~1 GFLOP): optimized for the memory path.
//  - 4 pixels/thread => b128 coalesced NT loads/stores on all 27 channel streams
//  - non-temporal hints keep init_dem resident in 192MB L2 for the gathers
//  - global_prefetch_b8 warms the 3-row bilinear gather neighborhood
//  - BRANCHLESS gathers: clamp indices, zero the bilinear *weights* for OOB
//    rows/cols -> no exec-mask save/restore, gathers pipeline freely
//  - single-pass zero-mean-mask algebra: acc = S_wms - mean * S_ws

#define KS    3
#define K2    9
#define BATCH 4
#define IH    1024
#define IW    1024
#define PLANE (IH * IW)

typedef float v4f __attribute__((ext_vector_type(4)));

__device__ __forceinline__ int iclamp(int v, int lo, int hi) {
    int t = v < lo ? lo : v;          // lowers to v_med3_i32 / max+min
    return t > hi ? hi : t;
}

__device__ __forceinline__ float bilin(const float* __restrict__ img, float ys, float xs) {
    float y0f = floorf(ys);
    float x0f = floorf(xs);
    float wy1 = ys - y0f;
    float wx1 = xs - x0f;
    float wy0 = 1.0f - wy1;
    float wx0 = 1.0f - wx1;

    int y0 = (int)y0f, x0 = (int)x0f;
    int y1 = y0 + 1,   x1 = x0 + 1;

    // Zero-padding outside the image: zero the row/col weights (branchless).
    wy0 = ((unsigned)y0 < (unsigned)IH) ? wy0 : 0.0f;
    wy1 = ((unsigned)y1 < (unsigned)IH) ? wy1 : 0.0f;
    wx0 = ((unsigned)x0 < (unsigned)IW) ? wx0 : 0.0f;
    wx1 = ((unsigned)x1 < (unsigned)IW) ? wx1 : 0.0f;

    // Clamped addresses -> all 4 gathers issue unconditionally.
    int yc0 = iclamp(y0, 0, IH - 1);
    int yc1 = iclamp(y1, 0, IH - 1);
    int xc0 = iclamp(x0, 0, IW - 1);
    int xc1 = iclamp(x1, 0, IW - 1);

    const float* __restrict__ r0 = img + (yc0 << 10);
    const float* __restrict__ r1 = img + (yc1 << 10);
    float v00 = r0[xc0];
    float v01 = r0[xc1];
    float v10 = r1[xc0];
    float v11 = r1[xc1];

    return wy0 * (wx0 * v00 + wx1 * v01) + wy1 * (wx0 * v10 + wx1 * v11);
}

__global__ __launch_bounds__(256)
void PostProcessor_44504451121717_kernel(const float* __restrict__ init,
                                         const float* __restrict__ wgt,
                                         const float* __restrict__ off,
                                         const float* __restrict__ wtap,
                                         const float* __restrict__ bias,
                                         float* __restrict__ out) {
    const int tid = blockIdx.x * 256 + (int)threadIdx.x;   // 1,048,576 threads total
    const int w4  = tid & 255;                             // 256 quads along W
    const int h   = (tid >> 8) & (IH - 1);
    const int b   = tid >> 18;
    const int x   = w4 << 2;                               // first of 4 pixels

    const float* __restrict__ img = init + (size_t)b * PLANE;
    const size_t rowoff = (size_t)h * IW + x;
    const float* __restrict__ wbase = wgt + (size_t)b * K2 * PLANE + rowoff;
    const float* __restrict__ obase = off + (size_t)b * 2 * K2 * PLANE + rowoff;

    // Warm the gather neighborhood (gfx1250 global_prefetch_b8). Branchless:
    // clamp the row instead of guarding, so no exec-mask churn here either.
#pragma unroll
    for (int r = -1; r <= 1; ++r) {
        int yy = iclamp(h + r, 0, IH - 1);
        __builtin_prefetch(img + (size_t)(yy << 10) + x, 0, 3);
    }

    // Tiny uniform params (9 tap weights + bias) — broadcast from cache.
    float wk[K2];
#pragma unroll
    for (int k = 0; k < K2; ++k) wk[k] = wtap[k];
    const float b0 = bias[0];

    v4f sumW  = {0.f, 0.f, 0.f, 0.f};  // Σ wgt_k            (for the tap mean)
    v4f accWS = {0.f, 0.f, 0.f, 0.f};  // Σ w_k * wgt_k * s_k
    v4f accS  = {0.f, 0.f, 0.f, 0.f};  // Σ w_k * s_k

#pragma unroll
    for (int k = 0; k < K2; ++k) {
        const int ky = k / KS - 1;     // tap row offset  (PAD=1)
        const int kx = k % KS - 1;     // tap col offset

        // Single-use 151/302MB streams: non-temporal b128 loads.
        v4f wv = __builtin_nontemporal_load((const v4f*)(wbase + (size_t)k * PLANE));
        v4f dy = __builtin_nontemporal_load((const v4f*)(obase + (size_t)(2 * k) * PLANE));
        v4f dx = __builtin_nontemporal_load((const v4f*)(obase + (size_t)(2 * k + 1) * PLANE));

#pragma unroll
        for (int j = 0; j < 4; ++j) {
            float ys = (float)(h + ky) + dy[j];
            float xs = (float)(x + j + kx) + dx[j];
            float s  = bilin(img, ys, xs);
            sumW[j] += wv[j];
            accWS[j] = fmaf(wk[k] * wv[j], s, accWS[j]);
            accS[j]  = fmaf(wk[k],          s, accS[j]);
        }
    }

    // refined = Σ w_k*s_k*(wgt_k - mean) + b + 1.0*init
    //         = accWS - (sumW/9)*accS + b + init
    v4f iv = *(const v4f*)(img + rowoff);   // residual read (cache-resident)
    v4f res;
    const float inv9 = 1.0f / 9.0f;
#pragma unroll
    for (int j = 0; j < 4; ++j)
        res[j] = accWS[j] - (sumW[j] * inv9) * accS[j] + b0 + iv[j];

    __builtin_nontemporal_store(res, (v4f*)(out + (size_t)b * PLANE + rowoff));
}

extern "C" void kernel_launch(void* const* d_in, const int* in_sizes, int n_in,
                              void* d_out, int out_size, void* d_ws, size_t ws_size,
                              hipStream_t stream) {
    const float* init = (const float*)d_in[0];   // [4,1,1024,1024]
    const float* wgt  = (const float*)d_in[1];   // [4,9,1024,1024]
    const float* off  = (const float*)d_in[2];   // [4,18,1024,1024]
    const float* wtap = (const float*)d_in[3];   // [1,1,3,3] -> 9 floats
    const float* bias = (const float*)d_in[4];   // [1]
    float* out = (float*)d_out;                  // [4,1,1024,1024]

    const int total_threads = BATCH * IH * (IW / 4);   // 1,048,576
    dim3 grid(total_threads / 256);                    // 4096 blocks
    dim3 block(256);                                   // 8 wave32s
    hipLaunchKernelGGL(PostProcessor_44504451121717_kernel, grid, block, 0, stream,
                       init, wgt, off, wtap, bias, out);
}